// get_loss_49950469653255
// MI455X (gfx1250) — compile-verified
//
#include <hip/hip_runtime.h>

typedef float v2f __attribute__((ext_vector_type(2)));
typedef float v8f __attribute__((ext_vector_type(8)));

#define NPTS  4096
#define TILES 256   // NPTS / 16

// One wave (32 threads) per block. Each block owns one 16-query tile of one
// batch and scans all 4096 targets in chunks of 16 using
// V_WMMA_F32_16X16X4_F32.
//
// The K=4 slot folds the affine terms into the matrix op:
//   A row  = [-2*q0, -2*q1, -2*q2, 1]
//   B col  = [ t0,    t1,    t2,  |t|^2]
//   D      = |t|^2 - 2*q.t        (squared distance minus per-row |q|^2)
// |q|^2 is constant per row, so it is added after the min reduction; sqrt is
// monotonic, so it is applied once per row at the end.
__global__ __launch_bounds__(32)
void chamfer_dir_kernel(const float* __restrict__ q,   // queries [4,4096,3]
                        const float* __restrict__ t,   // targets [4,4096,3]
                        float* __restrict__ partial)   // [4,256] per-tile sum of per-query min dists
{
    const int tile = blockIdx.x;    // 0..255
    const int b    = blockIdx.y;    // 0..3
    const int lane = threadIdx.x;   // 0..31
    const int col  = lane & 15;
    const int half = lane >> 4;

    // ---- load the 16-query tile into WMMA A layout (16x4 f32) ----
    // A layout (ISA 7.12.2): lanes 0-15 -> {K=0, K=1}, lanes 16-31 -> {K=2, K=3}
    const float* qb = q + ((size_t)b * NPTS + (size_t)tile * 16) * 3;
    float q0 = qb[col * 3 + 0];
    float q1 = qb[col * 3 + 1];
    float q2 = qb[col * 3 + 2];
    float qn = q0 * q0 + q1 * q1 + q2 * q2;   // |query[col]|^2

    v2f a;
    a.x = half ? (-2.0f * q2) : (-2.0f * q0);
    a.y = half ? 1.0f         : (-2.0f * q1);

    // D register r holds row M = r + 8*half; fetch those query norms via shuffles
    float qn8[8];
    #pragma unroll
    for (int r = 0; r < 8; ++r)
        qn8[r] = __shfl(qn, r + 8 * half, 32);

    float rmin[8];
    #pragma unroll
    for (int r = 0; r < 8; ++r) rmin[r] = 3.0e38f;

    const float* tb = t + (size_t)b * NPTS * 3;

    #pragma unroll 2
    for (int c = 0; c < TILES; ++c) {
        // ---- load 16 targets into WMMA B layout (4x16 f32) ----
        // B layout mirrors A: lanes 0-15 -> rows {K=0, K=1}, lanes 16-31 -> {K=2, K=3}
        const float* tp = tb + (size_t)(c * 16 + col) * 3;
        float t0 = tp[0], t1 = tp[1], t2 = tp[2];
        float tn = t0 * t0 + t1 * t1 + t2 * t2;   // |target[col]|^2

        v2f bmat;
        bmat.x = half ? t2 : t0;
        bmat.y = half ? tn : t1;

        v8f acc = {};
        // (neg_a, A, neg_b, B, c_mod, C, reuse_a, reuse_b)
        acc = __builtin_amdgcn_wmma_f32_16x16x4_f32(false, a, false, bmat,
                                                    (short)0, acc, false, false);

        // acc[r] = |t|^2 - 2*q.t  -> min directly (|q|^2 added later)
        #pragma unroll
        for (int r = 0; r < 8; ++r)
            rmin[r] = fminf(rmin[r], acc[r]);
    }

    // min across the 16 columns (lanes within each half), per row
    #pragma unroll
    for (int m = 1; m < 16; m <<= 1) {
        #pragma unroll
        for (int r = 0; r < 8; ++r)
            rmin[r] = fminf(rmin[r], __shfl_xor(rmin[r], m, 32));
    }

    // restore |q|^2, clamp, one sqrt per row (monotonicity), then sum rows
    float s = 0.0f;
    #pragma unroll
    for (int r = 0; r < 8; ++r)
        s += sqrtf(fmaxf(qn8[r] + rmin[r], 0.0f));
    s += __shfl_xor(s, 16, 32);   // fold the other half's 8 rows

    if (lane == 0) partial[b * TILES + tile] = s;
}

// Deterministic fixed-order reduction: out[b] = (sum dir1 + sum dir2) / 4096
__global__ __launch_bounds__(32)
void chamfer_reduce_kernel(const float* __restrict__ p1,
                           const float* __restrict__ p2,
                           float* __restrict__ out)
{
    const int b = blockIdx.x;
    if (threadIdx.x == 0) {
        float s1 = 0.0f;
        for (int i = 0; i < TILES; ++i) s1 += p1[b * TILES + i];
        float s2 = 0.0f;
        for (int i = 0; i < TILES; ++i) s2 += p2[b * TILES + i];
        out[b] = (s1 + s2) * (1.0f / NPTS);
    }
}

extern "C" void kernel_launch(void* const* d_in, const int* in_sizes, int n_in,
                              void* d_out, int out_size, void* d_ws, size_t ws_size,
                              hipStream_t stream) {
    const float* x = (const float*)d_in[0];   // [4,4096,3]
    const float* y = (const float*)d_in[1];   // [4,4096,3]
    float* out = (float*)d_out;               // [4]

    float* p1 = (float*)d_ws;                 // [4*256]
    float* p2 = p1 + 4 * TILES;               // [4*256]

    dim3 grid(TILES, 4);
    // dis1: for each x point, min distance to y set
    chamfer_dir_kernel<<<grid, 32, 0, stream>>>(x, y, p1);
    // dis2: for each y point, min distance to x set
    chamfer_dir_kernel<<<grid, 32, 0, stream>>>(y, x, p2);
    chamfer_reduce_kernel<<<4, 32, 0, stream>>>(p1, p2, out);
}